// BaseModel_77893526880336
// MI455X (gfx1250) — compile-verified
//
#include <hip/hip_runtime.h>
#include <math.h>

#define N_NODES 2048
#define DIM     64
#define NBINS   64
#define NPAIRS  16384
#define NEVENTS 262144
#define NC      (N_NODES * DIM)   // 131072 columns (node*dim)
#define EPSF    1e-6f

typedef __attribute__((ext_vector_type(2))) float v2f;
typedef __attribute__((ext_vector_type(8))) float v8f;

// workspace layout (bytes)
#define WS_BOUNDS 0       // 65 floats
#define WS_WIDTHS 512     // 64 floats
#define WS_ACC    1024    // 1 double
#define WS_VC     2048    // 65 * NC floats  (~34 MB)

__device__ __forceinline__ float wave_sum(float x) {
#pragma unroll
  for (int m = 16; m > 0; m >>= 1) x += __shfl_xor(x, m, 32);
  return x;
}

// ---------------- setup: softmax -> bounds / widths, zero accumulator -------
__global__ void setup_kernel(const float* __restrict__ bins_rwidth,
                             float* __restrict__ bounds,
                             float* __restrict__ widths,
                             double* __restrict__ acc) {
  if (threadIdx.x == 0) {
    float mx = -1e30f;
    for (int b = 0; b < NBINS; ++b) mx = fmaxf(mx, bins_rwidth[b]);
    float e[NBINS]; float s = 0.f;
    for (int b = 0; b < NBINS; ++b) { e[b] = expf(bins_rwidth[b] - mx); s += e[b]; }
    float inv = 1.f / s;
    float c = 0.f;
    bounds[0] = 0.f;
    for (int b = 0; b < NBINS; ++b) {
      float prev = c;
      c += e[b] * inv;              // * LAST_TIME (=1.0)
      bounds[b + 1] = c;
      widths[b] = c - prev;
    }
    *acc = 0.0;
  }
}

// ---------------- zero row 0 of Vc ------------------------------------------
__global__ void zero_vc0_kernel(float* __restrict__ vc) {
  int idx = blockIdx.x * blockDim.x + threadIdx.x;
  vc[idx] = 0.f;
}

// ---------------- Vc = L(widths, lower-tri) @ v  via V_WMMA_F32_16X16X4_F32 -
// Vc[k][c] = sum_{b<k} widths[b] * v[b][c], rows k=1..64 computed here.
__global__ void __launch_bounds__(256) vc_gemm_kernel(
    const float* __restrict__ v, const float* __restrict__ widths,
    float* __restrict__ vc) {
  __shared__ float s_w[NBINS];
  if (threadIdx.x < NBINS) s_w[threadIdx.x] = widths[threadIdx.x];
  __syncthreads();

  int wave  = blockIdx.x * (blockDim.x >> 5) + (threadIdx.x >> 5);
  int lane  = threadIdx.x & 31;
  int rtile = wave & 3;        // 4 row tiles cover output rows m=0..63 (Vc rows 1..64)
  int ctile = wave >> 2;       // 8192 column tiles of 16
  int half  = lane >> 4;       // lane half selects K pair / M group
  int l16   = lane & 15;
  int col   = ctile * 16 + l16;
  int rowbase = rtile * 16;    // Vc row = rowbase + m + 1
  int rowlim  = rowbase + l16; // A mask: include bin b iff b <= rowlim

  v8f c = {};
#pragma unroll
  for (int kk = 0; kk < NBINS; kk += 4) {
    int k0 = kk + 2 * half;
    // unconditional LDS reads (speculation safe) -> mask via v_cndmask, no
    // divergent predicated ds_load
    float w0 = s_w[k0];
    float w1 = s_w[k0 + 1];
    v2f a, b;
    a.x = (k0     <= rowlim) ? w0 : 0.f;
    a.y = (k0 + 1 <= rowlim) ? w1 : 0.f;
    b.x = v[(size_t)k0       * NC + col];
    b.y = v[(size_t)(k0 + 1) * NC + col];
    c = __builtin_amdgcn_wmma_f32_16x16x4_f32(false, a, false, b,
                                              (short)0, c, false, false);
  }
#pragma unroll
  for (int q = 0; q < 8; ++q) {
    int m = q + 8 * half;
    vc[(size_t)(rowbase + m + 1) * NC + col] = c[q];
  }
}

// ---------------- integral over pairs: one wave per pair --------------------
__global__ void __launch_bounds__(256) integral_kernel(
    const float* __restrict__ x0, const float* __restrict__ v,
    const float* __restrict__ beta, const int* __restrict__ node_pairs,
    const float* __restrict__ vc, double* __restrict__ acc) {
  int wave = blockIdx.x * (blockDim.x >> 5) + (threadIdx.x >> 5);
  int lane = threadIdx.x & 31;
  int p = wave;
  int i = node_pairs[p];
  int j = node_pairs[NPAIRS + p];
  float bsum = beta[i] + beta[j];

  size_t oi = (size_t)i * DIM + 2 * lane;
  size_t oj = (size_t)j * DIM + 2 * lane;
  float2 xa = *(const float2*)(x0 + oi);
  float2 xb = *(const float2*)(x0 + oj);
  float2 dx; dx.x = xa.x - xb.x; dx.y = xa.y - xb.y;

  float2 xt_prev = make_float2(0.f, 0.f);
  float numer_prev = 0.f;
  float local = 0.f;

  for (int k = 0; k <= NBINS; ++k) {
    float2 vci = *(const float2*)(vc + (size_t)k * NC + oi);
    float2 vcj = *(const float2*)(vc + (size_t)k * NC + oj);
    float2 xt; xt.x = dx.x + vci.x - vcj.x; xt.y = dx.y + vci.y - vcj.y;
    float n2 = xt.x * xt.x + xt.y * xt.y;
    float d1 = 0.f, d0 = 0.f;
    if (k > 0) {
      float2 vi = *(const float2*)(v + (size_t)(k - 1) * NC + oi);
      float2 vj = *(const float2*)(v + (size_t)(k - 1) * NC + oj);
      float2 dv; dv.x = vi.x - vj.x; dv.y = vi.y - vj.y;
      d1 = xt.x * dv.x + xt.y * dv.y;
      d0 = xt_prev.x * dv.x + xt_prev.y * dv.y;
    }
    n2 = wave_sum(n2);
    d1 = wave_sum(d1);
    d0 = wave_sum(d0);
    float nrm   = sqrtf(n2);
    float numer = nrm * expf(bsum - nrm);
    if (k > 0) local += numer / (d1 + EPSF) - numer_prev / (d0 + EPSF);
    xt_prev = xt;
    numer_prev = numer;
  }
  if (lane == 0) atomicAdd(acc, (double)local);
}

// ---------------- events: thread per event ----------------------------------
__global__ void __launch_bounds__(256) event_kernel(
    const float* __restrict__ x0, const float* __restrict__ v,
    const float* __restrict__ beta, const int* __restrict__ node_pairs,
    const float* __restrict__ event_times, const int* __restrict__ event_pair_idx,
    const float* __restrict__ bounds, const float* __restrict__ vc,
    double* __restrict__ acc) {
  __shared__ float s_b[NBINS + 1];
  if (threadIdx.x <= NBINS) s_b[threadIdx.x] = bounds[threadIdx.x];
  __syncthreads();

  int e = blockIdx.x * blockDim.x + threadIdx.x;
  float t = event_times[e];
  // searchsorted(inner=bounds[1..63], t, right): first m in [1,64) with bounds[m] > t
  int lo = 1, hi = NBINS;
  while (lo < hi) { int mid = (lo + hi) >> 1; if (s_b[mid] <= t) lo = mid + 1; else hi = mid; }
  int k = lo - 1;                     // bin index in [0,63]
  float rem = t - s_b[k];

  int pid = event_pair_idx[e];
  int i = node_pairs[pid];
  int j = node_pairs[NPAIRS + pid];
  const float4* x0i = (const float4*)(x0 + (size_t)i * DIM);
  const float4* x0j = (const float4*)(x0 + (size_t)j * DIM);
  const float4* vci = (const float4*)(vc + (size_t)k * NC + (size_t)i * DIM);
  const float4* vcj = (const float4*)(vc + (size_t)k * NC + (size_t)j * DIM);
  const float4* vi  = (const float4*)(v  + (size_t)k * NC + (size_t)i * DIM);
  const float4* vj  = (const float4*)(v  + (size_t)k * NC + (size_t)j * DIM);

  float s = 0.f;
#pragma unroll 4
  for (int c = 0; c < DIM / 4; ++c) {
    float4 a = x0i[c], b = x0j[c], ci = vci[c], cj = vcj[c], pi = vi[c], pj = vj[c];
    float xx = (a.x - b.x) + (ci.x - cj.x) + (pi.x - pj.x) * rem;
    float yy = (a.y - b.y) + (ci.y - cj.y) + (pi.y - pj.y) * rem;
    float zz = (a.z - b.z) + (ci.z - cj.z) + (pi.z - pj.z) * rem;
    float ww = (a.w - b.w) + (ci.w - cj.w) + (pi.w - pj.w) * rem;
    s += xx * xx + yy * yy + zz * zz + ww * ww;
  }
  float dist = sqrtf(s);
  // result = integral - sum(bsum - dist)  =>  add (dist - bsum)
  float contrib = dist - (beta[i] + beta[j]);
  contrib = wave_sum(contrib);
  if ((threadIdx.x & 31) == 0) atomicAdd(acc, (double)contrib);
}

// ---------------- finalize ---------------------------------------------------
__global__ void finalize_kernel(const double* __restrict__ acc, float* __restrict__ out) {
  if (threadIdx.x == 0 && blockIdx.x == 0) out[0] = (float)(*acc);
}

extern "C" void kernel_launch(void* const* d_in, const int* in_sizes, int n_in,
                              void* d_out, int out_size, void* d_ws, size_t ws_size,
                              hipStream_t stream) {
  const float* x0             = (const float*)d_in[0];
  const float* v              = (const float*)d_in[1];
  const float* beta           = (const float*)d_in[2];
  const float* bins_rwidth    = (const float*)d_in[3];
  const float* event_times    = (const float*)d_in[4];
  const int*   node_pairs     = (const int*)d_in[5];
  const int*   event_pair_idx = (const int*)d_in[6];
  float* out = (float*)d_out;

  char* ws = (char*)d_ws;
  float*  bounds = (float*)(ws + WS_BOUNDS);
  float*  widths = (float*)(ws + WS_WIDTHS);
  double* acc    = (double*)(ws + WS_ACC);
  float*  vc     = (float*)(ws + WS_VC);

  setup_kernel<<<1, 64, 0, stream>>>(bins_rwidth, bounds, widths, acc);
  zero_vc0_kernel<<<NC / 256, 256, 0, stream>>>(vc);
  // 4 row tiles x 8192 col tiles = 32768 waves, 8 waves/block
  vc_gemm_kernel<<<(4 * 8192) / 8, 256, 0, stream>>>(v, widths, vc);
  integral_kernel<<<NPAIRS / 8, 256, 0, stream>>>(x0, v, beta, node_pairs, vc, acc);
  event_kernel<<<NEVENTS / 256, 256, 0, stream>>>(x0, v, beta, node_pairs,
                                                  event_times, event_pair_idx,
                                                  bounds, vc, acc);
  finalize_kernel<<<1, 1, 0, stream>>>(acc, out);
}